// Coregistration_38208029065667
// MI455X (gfx1250) — compile-verified
//
#include <hip/hip_runtime.h>

// ---------------- problem constants ----------------
#define HH 512
#define WW 512
#define CS 8
#define HC 506                 // cropped size (512 - 6)
#define NPIX (HC * HC)
#define KW 41                  // MTF kernel size
#define KR 20                  // MTF radius
#define TK 56                  // Toeplitz K extent (16 outputs + 40 taps)
#define XT 4                   // output tiles along x per wave (64 cols)
#define TCOLS (XT * 16 + 40)   // 104 input cols needed
#define TSTRIDE 106            // padded LDS row stride (even -> 8B aligned)
#define KSTRIDE 72             // padded kernel row: 15 zeros + 41 taps + 16 zeros

typedef __attribute__((ext_vector_type(2))) float v2f;
typedef __attribute__((ext_vector_type(8))) float v8f;

// 12 nonzero taps of the half-pixel interpolator (_KX12; 13th tap is 0).
// Cross-correlation (no flip), SAME pad: out[x] = sum_t h[t]*in[x-6+t]
__constant__ float HPF[12] = {
    -1.20162964e-04f,  1.615524292e-03f, -1.0385513306e-02f,
     4.3619155884e-02f, -1.45397186478e-01f, 6.1066818237e-01f,
     6.1066818237e-01f, -1.45397186478e-01f, 4.3619155884e-02f,
    -1.0385513306e-02f, 1.615524292e-03f, -1.20162964e-04f };

__device__ __forceinline__ float ldz(const float* __restrict__ b, int y, int x,
                                     int H, int W) {
    return ((unsigned)y < (unsigned)H && (unsigned)x < (unsigned)W)
               ? b[y * W + x] : 0.f;
}

// ====== 41x41 depthwise MTF conv via V_WMMA_F32_16X16X4_F32 ================
// One wave computes a 16x64 output region as four 16x16 WMMA tiles:
//   D_j = sum_ky A_j,ky(16x56) x B_ky(56x16)
//   A_j,ky[m][k] = panE[y0+m+ky-20][x0+16j+k-20]   (edge-clamped, LDS slab)
//   B_ky[k][n]   = kpad[ky][15+k-n]                (zero-padded Toeplitz band)
// B is shared by all four tiles; all LDS loads are unconditional.
__global__ __launch_bounds__(32) void k_mtf_wmma(
    const float* __restrict__ pan, const float* __restrict__ mtf,
    float* __restrict__ P) {
    __shared__ float tile[TK][TSTRIDE];
    __shared__ float kpad[KW][KSTRIDE];

    const int lane = threadIdx.x;
    const int x0 = blockIdx.x * (XT * 16), y0 = blockIdx.y * 16, c = blockIdx.z;
    const float* pc = pan + c * HH * WW;

    for (int idx = lane; idx < TK * TCOLS; idx += 32) {
        int i = idx / TCOLS, j = idx % TCOLS;
        int yy = y0 + i - KR; yy = yy < 0 ? 0 : (yy > HH - 1 ? HH - 1 : yy);
        int xx = x0 + j - KR; xx = xx < 0 ? 0 : (xx > WW - 1 ? WW - 1 : xx);
        tile[i][j] = pc[yy * WW + xx];
    }
    for (int idx = lane; idx < KW * KSTRIDE; idx += 32) {
        int ky = idx / KSTRIDE, j = idx % KSTRIDE;
        int d = j - 15;
        kpad[ky][j] = ((unsigned)d <= 40u) ? mtf[c * KW * KW + ky * KW + d] : 0.f;
    }
    __syncthreads();

    v8f acc0 = {0.f,0.f,0.f,0.f,0.f,0.f,0.f,0.f};
    v8f acc1 = acc0, acc2 = acc0, acc3 = acc0;
    const int mn    = lane & 15;          // A row m == B col n
    const int khalf = (lane >> 4) << 1;   // K sub-offset: 0 or 2

    for (int ky = 0; ky < KW; ++ky) {
        const float* trow = &tile[mn + ky][0];
        const float* brow = &kpad[ky][15 - mn];
        for (int kk = 0; kk < TK / 4; ++kk) {
            const int k0 = kk * 4 + khalf;
            v2f B;  B.x  = brow[k0];      B.y  = brow[k0 + 1];
            v2f A0; A0.x = trow[k0];      A0.y = trow[k0 + 1];
            v2f A1; A1.x = trow[k0 + 16]; A1.y = trow[k0 + 17];
            v2f A2; A2.x = trow[k0 + 32]; A2.y = trow[k0 + 33];
            v2f A3; A3.x = trow[k0 + 48]; A3.y = trow[k0 + 49];
            acc0 = __builtin_amdgcn_wmma_f32_16x16x4_f32(
                false, A0, false, B, (short)0, acc0, false, false);
            acc1 = __builtin_amdgcn_wmma_f32_16x16x4_f32(
                false, A1, false, B, (short)0, acc1, false, false);
            acc2 = __builtin_amdgcn_wmma_f32_16x16x4_f32(
                false, A2, false, B, (short)0, acc2, false, false);
            acc3 = __builtin_amdgcn_wmma_f32_16x16x4_f32(
                false, A3, false, B, (short)0, acc3, false, false);
        }
    }
    float* oc = P + c * HH * WW;
    const int rhalf = (lane >> 4) * 8;
    for (int v = 0; v < 8; ++v) {
        int row = (y0 + v + rhalf) * WW;
        oc[row + x0 +      mn] = acc0[v];
        oc[row + x0 + 16 + mn] = acc1[v];
        oc[row + x0 + 32 + mn] = acc2[v];
        oc[row + x0 + 48 + mn] = acc3[v];
    }
}

// ================= vertical 12-tap half-pel prefilter (PV) =================
__global__ void k_conv_v(const float* __restrict__ P, float* __restrict__ PV) {
    int id = blockIdx.x * blockDim.x + threadIdx.x;
    int c = blockIdx.y;
    if (id >= HH * WW) return;
    int y = id / WW, x = id % WW;
    const float* pc = P + c * HH * WW;
    float s = 0.f;
#pragma unroll
    for (int t = 0; t < 12; ++t) s += HPF[t] * ldz(pc, y - 6 + t, x, HH, WW);
    PV[c * HH * WW + id] = s;
}

// ================= a-side (ms) stats: a = msc - box(msc)/256; sii ==========
__global__ void k_a1(const float* __restrict__ ms, float* __restrict__ S1) {
    int id = blockIdx.x * blockDim.x + threadIdx.x;
    int c = blockIdx.y;
    if (id >= NPIX) return;
    int y = id / HC, x = id % HC;
    const float* mc = ms + c * HH * WW;
    float s = 0.f;
    for (int dy = -7; dy <= 8; ++dy) {
        int j = y + dy;
        if ((unsigned)j < (unsigned)HC) s += mc[(j + 3) * WW + (x + 3)];
    }
    S1[c * NPIX + id] = s;
}

__global__ void k_a2(const float* __restrict__ ms, const float* __restrict__ S1,
                     float* __restrict__ A) {
    int id = blockIdx.x * blockDim.x + threadIdx.x;
    int c = blockIdx.y;
    if (id >= NPIX) return;
    int y = id / HC, x = id % HC;
    const float* s1c = S1 + c * NPIX;
    float s = 0.f;
    for (int dx = -7; dx <= 8; ++dx) s += ldz(s1c, y, x + dx, HC, HC);
    A[c * NPIX + id] = ms[c * HH * WW + (y + 3) * WW + (x + 3)] - s * (1.f / 256.f);
}

__global__ void k_a3(const float* __restrict__ A, float* __restrict__ S1) {
    int id = blockIdx.x * blockDim.x + threadIdx.x;
    int c = blockIdx.y;
    if (id >= NPIX) return;
    int y = id / HC, x = id % HC;
    const float* ac = A + c * NPIX;
    float s = 0.f;
    for (int dy = -7; dy <= 8; ++dy) {
        float v = ldz(ac, y + dy, x, HC, HC);
        s += v * v;
    }
    S1[c * NPIX + id] = s;
}

__global__ void k_a4(const float* __restrict__ S1, float* __restrict__ SII) {
    int id = blockIdx.x * blockDim.x + threadIdx.x;
    int c = blockIdx.y;
    if (id >= NPIX) return;
    int y = id / HC, x = id % HC;
    const float* s1c = S1 + c * NPIX;
    float s = 0.f;
    for (int dx = -7; dx <= 8; ++dx) s += ldz(s1c, y, x + dx, HC, HC);
    SII[c * NPIX + id] = fmaxf(s, 1e-20f);
}

// ================= per-shift: braw = crop(fineshift(p)) ====================
__global__ void k_braw(const float* __restrict__ P, const float* __restrict__ PV,
                       float* __restrict__ BR, int r_int, int c_int,
                       int use_v, int use_h) {
    int id = blockIdx.x * blockDim.x + threadIdx.x;
    int c = blockIdx.y;
    if (id >= NPIX) return;
    int y = id / HC, x = id % HC;
    const float* base = (use_v ? PV : P) + c * HH * WW;
    int v = 3 + y - r_int;
    int u = 3 + x - c_int;
    float s;
    if (use_h) {
        s = 0.f;
#pragma unroll
        for (int t = 0; t < 12; ++t) s += HPF[t] * ldz(base, v, u - 6 + t, HH, WW);
    } else {
        s = ldz(base, v, u, HH, WW);
    }
    BR[c * NPIX + id] = s;
}

// col-sum (16 wide) of braw
__global__ void k_colsum(const float* __restrict__ X, float* __restrict__ S) {
    int id = blockIdx.x * blockDim.x + threadIdx.x;
    int c = blockIdx.y;
    if (id >= NPIX) return;
    int y = id / HC, x = id % HC;
    const float* xc = X + c * NPIX;
    float s = 0.f;
    for (int dy = -7; dy <= 8; ++dy) s += ldz(xc, y + dy, x, HC, HC);
    S[c * NPIX + id] = s;
}

// row-sum of S, scaled -> mean_b
__global__ void k_rowsum_scale(const float* __restrict__ S, float* __restrict__ O,
                               float scale) {
    int id = blockIdx.x * blockDim.x + threadIdx.x;
    int c = blockIdx.y;
    if (id >= NPIX) return;
    int y = id / HC, x = id % HC;
    const float* sc = S + c * NPIX;
    float s = 0.f;
    for (int dx = -7; dx <= 8; ++dx) s += ldz(sc, y, x + dx, HC, HC);
    O[c * NPIX + id] = s * scale;
}

// col-sums of b*b and a*b, with b = braw - mean_b (zero outside crop)
__global__ void k_prod_colsum(const float* __restrict__ BR,
                              const float* __restrict__ MB,
                              const float* __restrict__ A,
                              float* __restrict__ C2, float* __restrict__ C3) {
    int id = blockIdx.x * blockDim.x + threadIdx.x;
    int c = blockIdx.y;
    if (id >= NPIX) return;
    int y = id / HC, x = id % HC;
    const float* brc = BR + c * NPIX;
    const float* mbc = MB + c * NPIX;
    const float* ac  = A  + c * NPIX;
    float s2 = 0.f, s3 = 0.f;
    for (int dy = -7; dy <= 8; ++dy) {
        int j = y + dy;
        if ((unsigned)j < (unsigned)HC) {
            float b = brc[j * HC + x] - mbc[j * HC + x];
            float a = ac[j * HC + x];
            s2 += b * b;
            s3 += a * b;
        }
    }
    C2[c * NPIX + id] = s2;
    C3[c * NPIX + id] = s3;
}

// finish boxes, form rho contributions, block-reduce, atomic accumulate
__global__ void k_rho(const float* __restrict__ C2, const float* __restrict__ C3,
                      const float* __restrict__ SII, float* __restrict__ rho,
                      int sidx) {
    __shared__ float red[256];
    int tid = threadIdx.x;
    int id = blockIdx.x * blockDim.x + tid;
    int c = blockIdx.y;
    float val = 0.f;
    if (id < NPIX) {
        int y = id / HC, x = id % HC;
        const float* c2 = C2 + c * NPIX;
        const float* c3 = C3 + c * NPIX;
        float sjj = 0.f, sij = 0.f;
        for (int dx = -7; dx <= 8; ++dx) {
            sjj += ldz(c2, y, x + dx, HC, HC);
            sij += ldz(c3, y, x + dx, HC, HC);
        }
        sjj = fmaxf(sjj, 1e-20f);
        float sii = SII[c * NPIX + id];
        val = sij / (sqrtf(sii * sjj) + 1e-20f);
    }
    red[tid] = val;
    __syncthreads();
    for (int off = 128; off > 0; off >>= 1) {
        if (tid < off) red[tid] += red[tid + off];
        __syncthreads();
    }
    if (tid == 0) atomicAdd(&rho[c * 81 + sidx], red[0]);
}

__global__ void k_zero(float* __restrict__ rho) {
    for (int i = threadIdx.x; i < CS * 81; i += blockDim.x) rho[i] = 0.f;
}

__global__ void k_argmax(const float* __restrict__ rho, int* __restrict__ out) {
    int c = threadIdx.x;
    if (c >= CS) return;
    float best = rho[c * 81];
    int bi = 0;
    for (int s = 1; s < 81; ++s) {
        float v = rho[c * 81 + s];
        if (v > best) { best = v; bi = s; }   // first-max tie-break (jnp.argmax)
    }
    out[c]      = bi / 9 - 4;  // r
    out[CS + c] = bi % 9 - 4;  // c
}

// =========================== host-side launcher ============================
extern "C" void kernel_launch(void* const* d_in, const int* in_sizes, int n_in,
                              void* d_out, int out_size, void* d_ws, size_t ws_size,
                              hipStream_t stream) {
    const float* ms  = (const float*)d_in[0];
    const float* pan = (const float*)d_in[1];
    const float* mtf = (const float*)d_in[2];

    float* ws = (float*)d_ws;
    const size_t SZP = (size_t)CS * HH * WW;   // 2,097,152
    const size_t SZC = (size_t)CS * NPIX;      // 2,048,288
    float* P   = ws;
    float* PV  = P   + SZP;
    float* A   = PV  + SZP;
    float* SII = A   + SZC;
    float* BR  = SII + SZC;
    float* S1  = BR  + SZC;
    float* MB  = S1  + SZC;
    float* C2  = MB  + SZC;
    float* C3  = C2  + SZC;
    float* RHO = C3  + SZC;                    // 8*81 floats

    // 1) MTF depthwise conv on the WMMA pipe (16x64 region per wave)
    dim3 gW(WW / (XT * 16), HH / 16, CS);
    k_mtf_wmma<<<gW, 32, 0, stream>>>(pan, mtf, P);

    // 2) vertical half-pel prefilter (used by shifts with odd m)
    dim3 gF((HH * WW + 255) / 256, CS);
    k_conv_v<<<gF, 256, 0, stream>>>(P, PV);

    // 3) a-side stats (independent of shift)
    dim3 gC((NPIX + 255) / 256, CS);
    k_a1<<<gC, 256, 0, stream>>>(ms, S1);
    k_a2<<<gC, 256, 0, stream>>>(ms, S1, A);
    k_a3<<<gC, 256, 0, stream>>>(A, S1);
    k_a4<<<gC, 256, 0, stream>>>(S1, SII);

    k_zero<<<1, 256, 0, stream>>>(RHO);

    // 4) 81 shifts
    for (int s = 0; s < 81; ++s) {
        int m = s / 9 - 4, n = s % 9 - 4;
        int mrem = ((m % 2) + 2) % 2;          // python-style remainder
        int nrem = ((n % 2) + 2) % 2;
        int r_int = (m - mrem) / 2;            // floor division
        int c_int = (n - nrem) / 2;
        k_braw<<<gC, 256, 0, stream>>>(P, PV, BR, r_int, c_int, mrem, nrem);
        k_colsum<<<gC, 256, 0, stream>>>(BR, S1);
        k_rowsum_scale<<<gC, 256, 0, stream>>>(S1, MB, 1.f / 256.f);
        k_prod_colsum<<<gC, 256, 0, stream>>>(BR, MB, A, C2, C3);
        k_rho<<<gC, 256, 0, stream>>>(C2, C3, SII, RHO, s);
    }

    // 5) argmax -> (r, c) int32, concatenated [r(8), c(8)]
    k_argmax<<<1, 32, 0, stream>>>(RHO, (int*)d_out);
}